// MultipleNegativesSymmetricRankingLoss_59760174956841
// MI455X (gfx1250) — compile-verified
//
#include <hip/hip_runtime.h>

// MI455X / gfx1250, wave32. fp32 WMMA path (V_WMMA_F32_16X16X4_F32).

typedef float v2f __attribute__((ext_vector_type(2)));
typedef float v8f __attribute__((ext_vector_type(8)));

#define B_SZ 4096
#define D_SZ 256
#define LDS_STRIDE 260   // 256 + 4 pad -> conflict-free ds_load_b64 across 16 lanes

// ---------------------------------------------------------------------------
// Kernel 1: row L2 norms for the 4 used matrices.  One wave per row.
// norms layout: [mat*4096 + row], mat: 0=anchor 1=positive 2=ref_text 3=ref_vision
// ---------------------------------------------------------------------------
__global__ __launch_bounds__(256) void mnr_norms(
    const float* __restrict__ A, const float* __restrict__ P,
    const float* __restrict__ T, const float* __restrict__ V,
    float* __restrict__ norms) {
  const int lane = threadIdx.x & 31;
  const int wid  = threadIdx.x >> 5;
  const int mat  = blockIdx.y;
  const float* X = (mat == 0) ? A : (mat == 1) ? P : (mat == 2) ? T : V;
  const int row  = blockIdx.x * 8 + wid;

  const float4* r4 = (const float4*)(X + (size_t)row * D_SZ);
  float4 u = r4[lane];        // cols [lane*4, lane*4+3]
  float4 w = r4[lane + 32];   // cols [128+lane*4, ...]
  float s = u.x * u.x + u.y * u.y + u.z * u.z + u.w * u.w
          + w.x * w.x + w.y * w.y + w.z * w.z + w.w * w.w;
  #pragma unroll
  for (int m = 1; m < 32; m <<= 1) s += __shfl_xor(s, m, 32);
  if (lane == 0) norms[mat * B_SZ + row] = sqrtf(s);
}

// ---------------------------------------------------------------------------
// Kernel 2: 8 directed jobs. Each wave owns a 16-row strip of the 4096x4096
// score matrix for one job; loops over 256 column tiles of 16 with LDS
// double-buffering (one __syncthreads per tile, global loads overlapped with
// the WMMA block). Per-tile 16x16 dot block via 64x v_wmma_f32_16x16x4_f32
// (K = 256) split across two independent accumulator chains.
// Accumulates sum(exp(s)) per row (no running max needed: |s| <= ~20) and
// the diagonal score; writes one partial  sum_rows(lse - s_ii)  per strip.
// ---------------------------------------------------------------------------
__global__ __launch_bounds__(256, 1) void mnr_main(
    const float* __restrict__ A, const float* __restrict__ P,
    const float* __restrict__ T, const float* __restrict__ V,
    const float* __restrict__ norms, float* __restrict__ partials) {
  __shared__ float ylds[2][16 * LDS_STRIDE];

  // Directed job table: (X rows, Y rows, norm idx X, norm idx Y)
  const int job = blockIdx.y;
  const float* X; const float* Y; int nx, ny;
  switch (job) {
    case 0:  X = A; nx = 0; Y = P; ny = 1; break;  // (anchor, positive)
    case 1:  X = P; nx = 1; Y = A; ny = 0; break;  // transpose of job 0
    case 2:  X = A; nx = 0; Y = T; ny = 2; break;  // (anchor, ref_text)
    case 3:  X = T; nx = 2; Y = A; ny = 0; break;
    case 4:  X = P; nx = 1; Y = T; ny = 2; break;  // (positive, ref_text)
    case 5:  X = T; nx = 2; Y = P; ny = 1; break;
    case 6:  X = P; nx = 1; Y = V; ny = 3; break;  // (positive, ref_vision)
    default: X = V; nx = 3; Y = P; ny = 1; break;
  }

  const int tid   = threadIdx.x;
  const int lane  = tid & 31;
  const int wid   = tid >> 5;
  const int strip = blockIdx.x * 8 + wid;   // 0..255, 16 rows each
  const int rbase = strip * 16;
  const int hi    = lane >> 4;              // lane half
  const int n     = lane & 15;              // column-within-tile / A-row index
  const int khalf = hi * 2;                 // ISA 16x4 f32 A layout: K offset

  // A fragments: lane holds X[rbase+n, kc*4+khalf .. +1] for 64 k-chunks.
  v2f afrag[64];
  const float* xrow = X + (size_t)(rbase + n) * D_SZ + khalf;
  #pragma unroll
  for (int kc = 0; kc < 64; ++kc)
    afrag[kc] = *(const v2f*)(xrow + kc * 4);

  // Row norms for this lane's 8 output rows (M = v + 8*hi).
  float rn[8];
  #pragma unroll
  for (int v = 0; v < 8; ++v)
    rn[v] = norms[nx * B_SZ + rbase + v + 8 * hi];

  float sumexp[8];
  #pragma unroll
  for (int v = 0; v < 8; ++v) sumexp[v] = 0.0f;
  float diag_acc = 0.0f;

  // Cooperative Y-tile staging coordinates (256 threads, 16x256 floats).
  const int yr = tid >> 4;
  const int yc = (tid & 15) * 16;

  // Prologue: stage tile 0 into buffer 0.
  {
    const float* ysrc = Y + (size_t)yr * D_SZ + yc;
    float* ydst = &ylds[0][yr * LDS_STRIDE + yc];
    #pragma unroll
    for (int i = 0; i < 4; ++i)
      *(float4*)(ydst + i * 4) = *(const float4*)(ysrc + i * 4);
  }

  int buf = 0;
  for (int ct = 0; ct < 256; ++ct) {
    __syncthreads();  // ylds[buf] ready; previous reads of ylds[buf^1] done

    // Kick off next tile's global loads (overlap with WMMA block below).
    const int nt = (ct + 1) & 255;  // wraps to a valid (redundant) tile
    const float* ysrc = Y + (size_t)(nt * 16 + yr) * D_SZ + yc;
    float4 nx0 = *(const float4*)(ysrc + 0);
    float4 nx1 = *(const float4*)(ysrc + 4);
    float4 nx2 = *(const float4*)(ysrc + 8);
    float4 nx3 = *(const float4*)(ysrc + 12);

    const float cn = norms[ny * B_SZ + ct * 16 + n];

    // 16x16 tile of raw dots: two independent WMMA accumulation chains.
    v8f acc0 = {};
    v8f acc1 = {};
    const float* brow = &ylds[buf][n * LDS_STRIDE + khalf];
    #pragma unroll
    for (int kc = 0; kc < 32; ++kc) {
      v2f b0 = *(const v2f*)(brow + (2 * kc) * 4);
      v2f b1 = *(const v2f*)(brow + (2 * kc + 1) * 4);
      acc0 = __builtin_amdgcn_wmma_f32_16x16x4_f32(
          false, afrag[2 * kc], false, b0, (short)0, acc0, false, false);
      acc1 = __builtin_amdgcn_wmma_f32_16x16x4_f32(
          false, afrag[2 * kc + 1], false, b1, (short)0, acc1, false, false);
    }

    // Store next tile into the alternate buffer (consumed after next barrier).
    {
      float* ydst = &ylds[buf ^ 1][yr * LDS_STRIDE + yc];
      *(float4*)(ydst + 0)  = nx0;
      *(float4*)(ydst + 4)  = nx1;
      *(float4*)(ydst + 8)  = nx2;
      *(float4*)(ydst + 12) = nx3;
    }

    const bool diag_tile = (ct == strip);
    #pragma unroll
    for (int v = 0; v < 8; ++v) {
      const float d = acc0[v] + acc1[v];
      const float denom = fmaxf(rn[v] * cn, 1e-6f);
      const float s = __fdividef(20.0f * d, denom);
      sumexp[v] += __expf(s);                     // |s| <= ~20: no max needed
      if (diag_tile && n == v + 8 * hi) diag_acc += s;
    }
    buf ^= 1;
  }

  // Row totals: reduce across the 16 lanes of each half, then lse = log(sum).
  float lse_sum = 0.0f;
  #pragma unroll
  for (int v = 0; v < 8; ++v) {
    float s = sumexp[v];
    s += __shfl_xor(s, 1, 32);
    s += __shfl_xor(s, 2, 32);
    s += __shfl_xor(s, 4, 32);
    s += __shfl_xor(s, 8, 32);
    lse_sum += __logf(s);
  }
  float contrib = ((n == 0) ? lse_sum : 0.0f) - diag_acc;
  contrib += __shfl_xor(contrib, 1, 32);
  contrib += __shfl_xor(contrib, 2, 32);
  contrib += __shfl_xor(contrib, 4, 32);
  contrib += __shfl_xor(contrib, 8, 32);
  contrib += __shfl_xor(contrib, 16, 32);
  if (lane == 0) partials[job * 256 + strip] = contrib;
}

// ---------------------------------------------------------------------------
// Kernel 3: deterministic reduction of 8*256 partials -> scalar loss.
// final = sum_j T_j / (4*B)
// ---------------------------------------------------------------------------
__global__ __launch_bounds__(256) void mnr_reduce(
    const float* __restrict__ partials, float* __restrict__ out) {
  __shared__ float sm[256];
  float s = 0.0f;
  for (int i = threadIdx.x; i < 2048; i += 256) s += partials[i];
  sm[threadIdx.x] = s;
  __syncthreads();
  for (int off = 128; off > 0; off >>= 1) {
    if ((int)threadIdx.x < off) sm[threadIdx.x] += sm[threadIdx.x + off];
    __syncthreads();
  }
  if (threadIdx.x == 0) out[0] = sm[0] * (1.0f / (4.0f * (float)B_SZ));
}

extern "C" void kernel_launch(void* const* d_in, const int* in_sizes, int n_in,
                              void* d_out, int out_size, void* d_ws, size_t ws_size,
                              hipStream_t stream) {
  (void)in_sizes; (void)n_in; (void)out_size; (void)ws_size;
  const float* A = (const float*)d_in[0];  // anchor
  const float* P = (const float*)d_in[1];  // positive
  // d_in[2] = reference_anchor: accepted but unused (matches reference)
  const float* T = (const float*)d_in[3];  // reference_positive_text
  const float* V = (const float*)d_in[4];  // reference_positive_vision

  float* norms    = (float*)d_ws;          // 4*4096 floats
  float* partials = norms + 4 * B_SZ;      // 8*256 floats

  mnr_norms <<<dim3(512, 4), 256, 0, stream>>>(A, P, T, V, norms);
  mnr_main  <<<dim3(32, 8),  256, 0, stream>>>(A, P, T, V, norms, partials);
  mnr_reduce<<<1,            256, 0, stream>>>(partials, (float*)d_out);
}